// LocalMessagePassing_70325794505395
// MI455X (gfx1250) — compile-verified
//
#include <hip/hip_runtime.h>
#include <math.h>

#define B_   4
#define N_   2048
#define D_   512
#define M_   (B_ * N_)        // 8192 rows
#define TOT  (M_ * D_)        // 4194304 elements

typedef float v2f __attribute__((ext_vector_type(2)));
typedef float v8f __attribute__((ext_vector_type(8)));
typedef int   v4i __attribute__((ext_vector_type(4)));

#if __has_builtin(__builtin_amdgcn_global_load_async_to_lds_b128)
#define HAVE_ASYNC_LDS 1
#endif

// builtin signature (from toolchain diagnostic): (v4i AS1*, v4i AS3*, int, imm int)
#define ASYNC_SRC(p) ((__attribute__((address_space(1))) v4i*)(void*)(p))
#define ASYNC_DST(p) ((__attribute__((address_space(3))) v4i*)(void*)(p))

// ---------------- device-global scratch (all rewritten every call) ---------------
__device__ float g_h [TOT];          // committed hidden state
__device__ float g_t1[TOT];          // scratch 1
__device__ float g_t2[TOT];          // scratch 2 (new_h / h_next)
__device__ float g_t3[TOT];          // scratch 3
__device__ float g_part[B_ * 8 * D_];// column partial sums for halt reduction
__device__ float g_scal[16];         // [0..3] remaining, [4..7] step weight, [8] active, [9] act_commit

__device__ __forceinline__ float gelu_exact(float x) {
    return 0.5f * x * (1.0f + erff(x * 0.70710678118654752440f));
}

__device__ __forceinline__ float* bufsel(int c) {
    switch (c) {
        case 0: return g_h;
        case 1: return g_t1;
        case 2: return g_t2;
        case 3: return g_t3;
    }
    return nullptr;
}

#define EPI_NONE  0
#define EPI_BIAS  1
#define EPI_MSG   2   // v = v / counts(row) + bias[n]
#define EPI_GELU  3   // v = gelu(v + bias[n])
#define EPI_RESID 4   // v = v + bias[n] + resid[m][n]

// ---------------------------------------------------------------------------------
// C[M,512] = A[M,K] @ W[K,512]  (A optionally "virtually concatenated" from two
// M x 512 buffers when K == 1024).  f32 WMMA 16x16x4, block tile 128x128,
// 8 waves, wave tile 32x64 (2x4 tiles of 16x16).
//   A tile in LDS:  row-major [r][k],  stride 36  (conflict-free b64 frag loads)
//   W tile in LDS:  TRANSPOSED [n][k], stride 36  (conflict-free b64 frag loads)
// A tile filled with GLOBAL_LOAD_ASYNC_TO_LDS_B128 when available (ASYNCcnt).
// ---------------------------------------------------------------------------------
__global__ __launch_bounds__(256)
void gemm_wmma(int acode, int abcode, const float* __restrict__ W, int ccode,
               const float* __restrict__ bias, int rescode, int K, int epi)
{
    __shared__ float ldsA [128 * 36];
    __shared__ float ldsWt[128 * 36];

    const float* Aa = bufsel(acode);
    const float* Ab = (abcode >= 0) ? bufsel(abcode) : nullptr;
    float*       C  = bufsel(ccode);
    const float* Rs = (rescode >= 0) ? bufsel(rescode) : nullptr;

    const int tid  = threadIdx.x;
    const int lane = tid & 31;
    const int wid  = tid >> 5;
    const int wm   = (wid & 3) * 32;   // wave M offset in block tile
    const int wn   = (wid >> 2) * 64;  // wave N offset in block tile
    const int m0   = blockIdx.x * 128;
    const int n0   = blockIdx.y * 128;
    const int lm   = lane & 15;
    const int lh   = lane >> 4;

    v8f acc[2][4];
    #pragma unroll
    for (int i = 0; i < 2; ++i)
        #pragma unroll
        for (int j = 0; j < 4; ++j)
            #pragma unroll
            for (int r = 0; r < 8; ++r) acc[i][j][r] = 0.0f;

    const int nkt = K >> 5;
    for (int kt = 0; kt < nkt; ++kt) {
        const int kk = kt << 5;
        const float* As = Aa;
        int kc = kk;
        if (Ab && kk >= 512) { As = Ab; kc = kk - 512; }

        // ---- W tile: k-strided global b32 loads into regs (coalesced over n) ----
        float rwv[4][4];
        #pragma unroll
        for (int j = 0; j < 4; ++j) {
            const int l  = tid + j * 256;      // 1024 (n, k-quad) slots
            const int n  = l & 127;
            const int k0 = (l >> 7) << 2;
            #pragma unroll
            for (int t = 0; t < 4; ++t)
                rwv[j][t] = W[(kk + k0 + t) * D_ + n0 + n];
        }
#ifndef HAVE_ASYNC_LDS
        float4 ra[4];
        #pragma unroll
        for (int j = 0; j < 4; ++j) {
            const int l = tid + j * 256;
            const int r = l >> 3, c4 = (l & 7) << 2;
            ra[j] = *(const float4*)&As[(m0 + r) * D_ + kc + c4];
        }
#endif
        __syncthreads();   // previous tile fully consumed

        // ---- A tile: async global -> LDS (per-lane b128), ASYNCcnt-tracked ------
#ifdef HAVE_ASYNC_LDS
        #pragma unroll
        for (int j = 0; j < 4; ++j) {
            const int l = tid + j * 256;       // 1024 float4 slots: 128 rows x 8
            const int r = l >> 3, c4 = (l & 7) << 2;
            __builtin_amdgcn_global_load_async_to_lds_b128(
                ASYNC_SRC(&As[(m0 + r) * D_ + kc + c4]),
                ASYNC_DST(&ldsA[r * 36 + c4]), 0, 0);
        }
#else
        #pragma unroll
        for (int j = 0; j < 4; ++j) {
            const int l = tid + j * 256;
            const int r = l >> 3, c4 = (l & 7) << 2;
            *(float4*)&ldsA[r * 36 + c4] = ra[j];
        }
#endif
        // ---- W tile: transposed LDS store (float4 along k, stride-36 over n) ----
        #pragma unroll
        for (int j = 0; j < 4; ++j) {
            const int l  = tid + j * 256;
            const int n  = l & 127;
            const int k0 = (l >> 7) << 2;
            *(float4*)&ldsWt[n * 36 + k0] =
                make_float4(rwv[j][0], rwv[j][1], rwv[j][2], rwv[j][3]);
        }
#ifdef HAVE_ASYNC_LDS
#if __has_builtin(__builtin_amdgcn_s_wait_asynccnt)
        __builtin_amdgcn_s_wait_asynccnt(0);
#else
        asm volatile("s_wait_asynccnt 0x0" ::: "memory");
#endif
#endif
        __syncthreads();

        // ---- 8 x (K=4) WMMA steps: both fragments are single aligned b64 loads --
        #pragma unroll
        for (int ks = 0; ks < 8; ++ks) {
            const int kb = (ks << 2) + (lh << 1);
            v2f af[2], bf[4];
            #pragma unroll
            for (int tM = 0; tM < 2; ++tM)
                af[tM] = *(const v2f*)&ldsA[(wm + tM * 16 + lm) * 36 + kb];
            #pragma unroll
            for (int tN = 0; tN < 4; ++tN)
                bf[tN] = *(const v2f*)&ldsWt[(wn + tN * 16 + lm) * 36 + kb];
            #pragma unroll
            for (int tM = 0; tM < 2; ++tM)
                #pragma unroll
                for (int tN = 0; tN < 4; ++tN)
                    acc[tM][tN] = __builtin_amdgcn_wmma_f32_16x16x4_f32(
                        false, af[tM], false, bf[tN], (short)0, acc[tM][tN],
                        false, false);
        }
    }

    // ---- fused epilogue + store ------------------------------------------------
    #pragma unroll
    for (int tM = 0; tM < 2; ++tM) {
        const int mb = m0 + wm + tM * 16 + (lh << 3);
        #pragma unroll
        for (int tN = 0; tN < 4; ++tN) {
            const int n = n0 + wn + tN * 16 + lm;
            #pragma unroll
            for (int r = 0; r < 8; ++r) {
                const int m = mb + r;
                float v = acc[tM][tN][r];
                if (epi == EPI_BIAS) {
                    v += bias[n];
                } else if (epi == EPI_MSG) {
                    const int i = m & (N_ - 1);
                    const float cnt = (float)((i + 1 < 5) ? (i + 1) : 5);
                    v = v / cnt + bias[n];
                } else if (epi == EPI_GELU) {
                    v = gelu_exact(v + bias[n]);
                } else if (epi == EPI_RESID) {
                    v = v + bias[n] + Rs[m * D_ + n];
                }
                C[m * D_ + n] = v;
            }
        }
    }
}

// ---------------------------------------------------------------------------------
// S[row] = sum_{d=0..min(i,4)} gelu(A1[row] + B1[row-d])  (A1 has msg_b1 folded in)
// ---------------------------------------------------------------------------------
__global__ __launch_bounds__(256)
void window_gelu_sum()
{
    const int row  = blockIdx.x;
    const int i    = row & (N_ - 1);
    const int dmax = (i < 4) ? i : 4;
    const float* A = &g_t1[row * D_];
    for (int c = threadIdx.x; c < D_; c += 256) {
        const float a = A[c];
        float s = 0.0f;
        for (int d = 0; d <= dmax; ++d)
            s += gelu_exact(a + g_t2[(row - d) * D_ + c]);
        g_t3[row * D_ + c] = s;
    }
}

// ---------------------------------------------------------------------------------
// Row LayerNorm in-place on g_t2 (one block per row, 256 threads x 2 elems).
// ---------------------------------------------------------------------------------
__global__ __launch_bounds__(256)
void ln_kernel(const float* __restrict__ gam, const float* __restrict__ bet)
{
    __shared__ float s1[256], s2[256];
    const int row = blockIdx.x;
    const int t   = threadIdx.x;
    float* p = &g_t2[row * D_];
    const float x0 = p[t], x1 = p[t + 256];
    s1[t] = x0 + x1;
    s2[t] = x0 * x0 + x1 * x1;
    __syncthreads();
    for (int o = 128; o > 0; o >>= 1) {
        if (t < o) { s1[t] += s1[t + o]; s2[t] += s2[t + o]; }
        __syncthreads();
    }
    const float mean = s1[0] * (1.0f / 512.0f);
    const float var  = s2[0] * (1.0f / 512.0f) - mean * mean;
    const float rstd = rsqrtf(var + 1e-5f);
    p[t]       = (x0 - mean) * rstd * gam[t]       + bet[t];
    p[t + 256] = (x1 - mean) * rstd * gam[t + 256] + bet[t + 256];
}

// ---------------------------------------------------------------------------------
// Column partial sums of h_next per batch: g_part[b][chunk][c]  (grid 4 x 8, 512 thr)
// ---------------------------------------------------------------------------------
__global__ __launch_bounds__(512)
void colsum_kernel()
{
    const int b = blockIdx.x, ch = blockIdx.y, t = threadIdx.x;
    float s = 0.0f;
    const int base = (b * N_ + ch * 256) * D_ + t;
    for (int r = 0; r < 256; ++r) s += g_t2[base + r * D_];
    g_part[(b * 8 + ch) * D_ + t] = s;
}

// ---------------------------------------------------------------------------------
// Per-step scalar bookkeeping: halt, step weights, remaining, kl, done/active.
// ---------------------------------------------------------------------------------
__global__ __launch_bounds__(512)
void scalar_step(int step, const float* __restrict__ haltW,
                 const float* __restrict__ haltb, float* __restrict__ klout)
{
    __shared__ float red[512];
    __shared__ float halt[B_];
    const int t = threadIdx.x;
    if (step >= 2 && step < 5) {
        for (int b = 0; b < B_; ++b) {
            float s = 0.0f;
            for (int j = 0; j < 8; ++j) s += g_part[(b * 8 + j) * D_ + t];
            red[t] = (s * (1.0f / (float)N_)) * haltW[t];
            __syncthreads();
            for (int o = 256; o > 0; o >>= 1) {
                if (t < o) red[t] += red[t + o];
                __syncthreads();
            }
            if (t == 0) halt[b] = 1.0f / (1.0f + expf(-(red[0] + haltb[0])));
            __syncthreads();
        }
    } else {
        if (t == 0) {
            const float hv = (step < 2) ? 0.0f : 1.0f;
            for (int b = 0; b < B_; ++b) halt[b] = hv;
        }
        __syncthreads();
    }
    if (t == 0) {
        const float act = g_scal[8];
        g_scal[9] = act;                       // gate used by this step's commit
        float hm = 0.0f;
        for (int b = 0; b < B_; ++b) {
            g_scal[4 + b] = g_scal[b] * halt[b];   // step_weight per batch
            hm += 0.25f * halt[b];                 // halt.mean()
        }
        if (act > 0.5f) {
            float mx = 0.0f;
            for (int b = 0; b < B_; ++b) {
                g_scal[b] *= (1.0f - halt[b]);
                mx = fmaxf(mx, g_scal[b]);
            }
            const float p = 0.2f * powf(0.8f, (float)step);
            klout[0] += (p * (logf(p) - logf(hm + 1e-8f))) * (1.0f / 6.0f);
            if (mx < 0.01f) g_scal[8] = 0.0f;      // done -> inactive next steps
        }
    }
}

// ---------------------------------------------------------------------------------
// Gated commit: output += w_b * h_next ; h = h_next  (only while active)
// ---------------------------------------------------------------------------------
__global__ __launch_bounds__(256)
void commit_kernel(float* __restrict__ out)
{
    const int i = blockIdx.x * 256 + threadIdx.x;
    const float act = g_scal[9];
    if (act > 0.5f) {
        const int b = i >> 20;                 // 2048*512 = 1<<20 per batch
        const float hn = g_t2[i];
        out[i] += g_scal[4 + b] * hn;
        g_h[i] = hn;
    }
}

__global__ __launch_bounds__(256)
void init_kernel(const float* __restrict__ summ, float* __restrict__ out)
{
    const int i = blockIdx.x * 256 + threadIdx.x;
    g_h[i] = summ[i];
    out[i] = 0.0f;
}

__global__ void init_scal(float* __restrict__ klout)
{
    if (threadIdx.x == 0) {
        for (int b = 0; b < B_; ++b) g_scal[b] = 1.0f;
        g_scal[8] = 1.0f;
        klout[0] = 0.0f;
    }
}

// ---------------------------------------------------------------------------------
extern "C" void kernel_launch(void* const* d_in, const int* in_sizes, int n_in,
                              void* d_out, int out_size, void* d_ws, size_t ws_size,
                              hipStream_t stream)
{
    const float* summ = (const float*)d_in[0];
    const float* mW1  = (const float*)d_in[1];   // (1024, 512)
    const float* mb1  = (const float*)d_in[2];
    const float* mW2  = (const float*)d_in[3];   // (512, 512)
    const float* mb2  = (const float*)d_in[4];
    const float* uW1  = (const float*)d_in[5];   // (1024, 512)
    const float* ub1  = (const float*)d_in[6];
    const float* uW2  = (const float*)d_in[7];   // (512, 512)
    const float* ub2  = (const float*)d_in[8];
    const float* hW   = (const float*)d_in[9];   // (512,)
    const float* hb   = (const float*)d_in[10];  // (1,)
    const float* lg   = (const float*)d_in[11];
    const float* lb   = (const float*)d_in[12];

    float* out   = (float*)d_out;
    float* klout = out + TOT;

    const dim3 gG(M_ / 128, D_ / 128);   // 64 x 4 blocks
    const dim3 bG(256);

    init_kernel<<<TOT / 256, 256, 0, stream>>>(summ, out);
    init_scal<<<1, 64, 0, stream>>>(klout);

    for (int step = 0; step < 6; ++step) {
        // msg layer 1 (split): t1 = h@W1_top + b1 ; t2 = h@W1_bot
        gemm_wmma<<<gG, bG, 0, stream>>>(0, -1, mW1,            1, mb1, -1, 512, EPI_BIAS);
        gemm_wmma<<<gG, bG, 0, stream>>>(0, -1, mW1 + 512 * D_, 2, nullptr, -1, 512, EPI_NONE);
        // t3 = sum_d gelu(t1_i + t2_{i-d})
        window_gelu_sum<<<M_, 256, 0, stream>>>();
        // msgs (t1) = t3@W2 / counts + b2
        gemm_wmma<<<gG, bG, 0, stream>>>(3, -1, mW2, 1, mb2, -1, 512, EPI_MSG);
        // G (t3) = gelu(concat(h, msgs)@U1 + b1)      (K = 1024, dual-A)
        gemm_wmma<<<gG, bG, 0, stream>>>(0, 1, uW1, 3, ub1, -1, 1024, EPI_GELU);
        // new_h (t2) = G@U2 + b2 + h
        gemm_wmma<<<gG, bG, 0, stream>>>(3, -1, uW2, 2, ub2, 0, 512, EPI_RESID);
        // h_next = LN(new_h), in place on t2
        ln_kernel<<<M_, 256, 0, stream>>>(lg, lb);
        // halting reduction (only needed on adaptive steps)
        if (step >= 2 && step < 5)
            colsum_kernel<<<dim3(B_, 8), 512, 0, stream>>>();
        scalar_step<<<1, 512, 0, stream>>>(step, hW, hb, klout);
        commit_kernel<<<TOT / 256, 256, 0, stream>>>(out);
    }
}